// End2End_49495203119139
// MI455X (gfx1250) — compile-verified
//
#include <hip/hip_runtime.h>
#include <hip/hip_bf16.h>

#define BATCH    16
#define MAX_LEN  128
#define VOCAB    32128
#define EMBED    768
#define NTOK     (BATCH * MAX_LEN)

typedef float v4f __attribute__((ext_vector_type(4)));

// ---------------------------------------------------------------------------
// Kernel A: per-token argmax over V of (logits + gumbel), then map through the
// grid_sample-nearest row index math (identity up to f32 rounding, replicated
// exactly). One block (8 wave32s) per token streams 257 KB with NT b128 loads.
// ---------------------------------------------------------------------------
__global__ __launch_bounds__(256)
void argmax_row_kernel(const float* __restrict__ logits,
                       const float* __restrict__ gumbel,
                       int* __restrict__ rowIdx) {
    const int t   = blockIdx.x;               // token id in [0, NTOK)
    const int tid = threadIdx.x;
    const size_t base = (size_t)t * VOCAB;

    const v4f* __restrict__ lg = (const v4f*)(logits + base);
    const v4f* __restrict__ gm = (const v4f*)(gumbel + base);
    const int nvec = VOCAB / 4;               // 8032, exact

    float bestV = -INFINITY;
    int   bestI = 0x7fffffff;

    for (int v = tid; v < nvec; v += 256) {
        // streamed once; keep out of L2 so the embedding table stays resident
        v4f a = __builtin_nontemporal_load(&lg[v]);
        v4f g = __builtin_nontemporal_load(&gm[v]);
        const int i0 = 4 * v;
        float s;
        s = a.x + g.x; if (s > bestV) { bestV = s; bestI = i0;     }
        s = a.y + g.y; if (s > bestV) { bestV = s; bestI = i0 + 1; }
        s = a.z + g.z; if (s > bestV) { bestV = s; bestI = i0 + 2; }
        s = a.w + g.w; if (s > bestV) { bestV = s; bestI = i0 + 3; }
    }

    __shared__ float sVal[256];
    __shared__ int   sIdx[256];
    sVal[tid] = bestV;
    sIdx[tid] = bestI;
    __syncthreads();

    // tree reduce; tiebreak = smallest index (jnp.argmax first occurrence)
    for (int off = 128; off > 0; off >>= 1) {
        if (tid < off) {
            float v2 = sVal[tid + off];
            int   i2 = sIdx[tid + off];
            if (v2 > sVal[tid] || (v2 == sVal[tid] && i2 < sIdx[tid])) {
                sVal[tid] = v2;
                sIdx[tid] = i2;
            }
        }
        __syncthreads();
    }

    if (tid == 0) {
        const int hot = sIdx[0];
        // replicate jnp f32 math: coord = linspace(-1,1,V)[hot]
        const float step  = __fdiv_rn(2.0f, (float)(VOCAB - 1));
        const float coord = __fadd_rn(-1.0f, __fmul_rn((float)hot, step));
        // _nearest_idx: ix = ((coord+1)*V - 1)/2 ; round-half-even; clip; cast
        const float ix = __fdiv_rn(
            __fsub_rn(__fmul_rn(__fadd_rn(coord, 1.0f), (float)VOCAB), 1.0f),
            2.0f);
        float r = rintf(ix);
        r = fminf(fmaxf(r, 0.0f), (float)(VOCAB - 1));
        rowIdx[t] = (int)r;
    }
}

// ---------------------------------------------------------------------------
// Kernel B: passage branch per batch row. roll(psg,1) with BOS=1, flipped
// mask, gather by (l - shifts) mod L, flag = (l >= first nonzero index).
// ---------------------------------------------------------------------------
__global__ __launch_bounds__(MAX_LEN)
void passage_kernel(const int* __restrict__ mask,
                    const int* __restrict__ psg_ids,
                    int* __restrict__ trunc,
                    int* __restrict__ flag) {
    const int b = blockIdx.x;
    const int l = threadIdx.x;

    __shared__ int m[MAX_LEN];
    __shared__ int extr[MAX_LEN];
    __shared__ int lenS;
    __shared__ int firstNz;

    m[l] = mask[b * MAX_LEN + l];
    if (l == 0) { lenS = 0; firstNz = MAX_LEN; }
    __syncthreads();

    atomicAdd(&lenS, m[l]);                       // shifts = sum(mask)

    const int psg     = (l == 0) ? 1 : psg_ids[b * MAX_LEN + l - 1];
    const int flipped = 1 - m[MAX_LEN - 1 - l];
    extr[l] = flipped * psg;
    __syncthreads();

    const int len = lenS;                         // in [0, 128]
    const int pos = (l - len + MAX_LEN) % MAX_LEN;
    const int tv  = extr[pos];
    trunc[b * MAX_LEN + l] = tv;
    if (tv != 0) atomicMin(&firstNz, l);
    __syncthreads();

    flag[b * MAX_LEN + l] = (l >= firstNz) ? 1 : 0;
}

// ---------------------------------------------------------------------------
// Kernel C: out[t,e] = mask[t]*W[row[t], col_idx[e]] + flag[t]*W[trunc[t], e].
// W (99 MB) fits in the 192 MB L2; rows are hot. col_idx replicated in f32.
// ---------------------------------------------------------------------------
__global__ __launch_bounds__(256)
void combine_kernel(const float* __restrict__ W,
                    const int* __restrict__ mask,
                    const int* __restrict__ rowIdx,
                    const int* __restrict__ trunc,
                    const int* __restrict__ flag,
                    float* __restrict__ out) {
    const int t   = blockIdx.x;
    const int tid = threadIdx.x;

    const float mk = (float)mask[t];
    const float* __restrict__ wrow = W + (size_t)rowIdx[t] * EMBED;
    const float* __restrict__ prow = W + (size_t)trunc[t] * EMBED;
    const int fl = flag[t];

    if (fl) {
        // gfx1250 global_prefetch_b8: warm the passage row through L2
        __builtin_prefetch(&prow[tid], 0, 1);
    }

    const float stepE = __fdiv_rn(2.0f, (float)(EMBED - 1));

    #pragma unroll
    for (int k = 0; k < EMBED / 256; ++k) {
        const int e = tid + k * 256;
        // col_idx = _nearest_idx(linspace(-1,1,E)[e], E), exact f32 replica
        const float coord = __fadd_rn(-1.0f, __fmul_rn((float)e, stepE));
        const float ix = __fdiv_rn(
            __fsub_rn(__fmul_rn(__fadd_rn(coord, 1.0f), (float)EMBED), 1.0f),
            2.0f);
        float rc = rintf(ix);
        rc = fminf(fmaxf(rc, 0.0f), (float)(EMBED - 1));
        const int col = (int)rc;

        float v = mk * wrow[col];
        if (fl) v += prow[e];
        out[(size_t)t * EMBED + e] = v;
    }
}

extern "C" void kernel_launch(void* const* d_in, const int* in_sizes, int n_in,
                              void* d_out, int out_size, void* d_ws, size_t ws_size,
                              hipStream_t stream) {
    const float* logits = (const float*)d_in[0];   // [16,128,32128] f32
    const int*   mask   = (const int*)  d_in[1];   // [16,128] i32
    const int*   psg    = (const int*)  d_in[2];   // [16,128] i32
    const float* W      = (const float*)d_in[3];   // [32128,768] f32
    const float* gumbel = (const float*)d_in[4];   // [16,128,32128] f32
    float*       out    = (float*)d_out;           // [16,128,768] f32

    int* rowIdx = (int*)d_ws;                      // NTOK ints
    int* trunc  = rowIdx + NTOK;                   // NTOK ints
    int* flag   = trunc + NTOK;                    // NTOK ints

    argmax_row_kernel<<<NTOK, 256, 0, stream>>>(logits, gumbel, rowIdx);
    passage_kernel<<<BATCH, MAX_LEN, 0, stream>>>(mask, psg, trunc, flag);
    combine_kernel<<<NTOK, 256, 0, stream>>>(W, mask, rowIdx, trunc, flag, out);
}